// TransducerLoss_18915035972188
// MI455X (gfx1250) — compile-verified
//
#include <hip/hip_runtime.h>
#include <hip/hip_bf16.h>
#include <stdint.h>
#include <stddef.h>

#define B_ 32
#define T_ 512
#define U_ 128
#define V_ 1024
#define EP 128   // emit row pitch (padded from 127 for alignment / bank behavior)

typedef unsigned int uint32x4 __attribute__((ext_vector_type(4)));
typedef int          int32x4  __attribute__((ext_vector_type(4)));
typedef int          int32x8  __attribute__((ext_vector_type(8)));

#if defined(__AMDGCN__) && __has_builtin(__builtin_amdgcn_tensor_load_to_lds) && __has_builtin(__builtin_amdgcn_s_wait_tensorcnt)
#define USE_TDM 1
#else
#define USE_TDM 0
#endif

#if USE_TDM
// DMA a strided column of `rows` f32 elements (element stride `stride_elems`)
// from global memory into `rows` consecutive f32 slots in LDS, via the
// CDNA5 Tensor Data Mover (tensor_load_to_lds, tracked by TENSORcnt).
__device__ __forceinline__ void tdm_load_col_f32(const float* gptr,
                                                 uint32_t lds_byte,
                                                 uint32_t rows,
                                                 uint32_t stride_elems) {
  uint64_t ga = (uint64_t)(uintptr_t)gptr;
  // ---- D# group 0: count=1, lds_addr, global_addr[56:0], type=2 ----
  uint32x4 g0;
  g0.x = 1u;                                           // count=1, user mode
  g0.y = lds_byte;                                     // LDS byte address
  g0.z = (uint32_t)ga;                                 // global_addr[31:0]
  g0.w = (uint32_t)((ga >> 32) & 0x01FFFFFFull)        // global_addr[56:32]
         | (2u << 30);                                 // type = 2 ("image")
  // ---- D# group 1 ----
  const uint32_t td0   = stride_elems;   // tensor_dim0 (x length, elements)
  const uint32_t td1   = rows;           // tensor_dim1
  const uint32_t tile0 = 1u;             // 1 element in x  -> column extraction
  const uint32_t tile1 = rows;           // `rows` lines in y
  const uint32_t tile2 = 0u;             // unused
  const uint64_t s0    = stride_elems;   // tensor_dim0_stride (elements)
  const uint64_t s1    = 0ull;
  int32x8 g1;
  g1[0] = (int)(2u << 16);                                               // data_size=4B; mask=0
  g1[1] = (int)((td0 & 0xFFFFu) << 16);                                  // abar=0 | tensor_dim0[15:0]
  g1[2] = (int)(((td0 >> 16) & 0xFFFFu) | ((td1 & 0xFFFFu) << 16));
  g1[3] = (int)(((td1 >> 16) & 0xFFFFu) | ((tile0 & 0xFFFFu) << 16));
  g1[4] = (int)((tile1 & 0xFFFFu) | ((tile2 & 0xFFFFu) << 16));
  g1[5] = (int)(uint32_t)(s0 & 0xFFFFFFFFull);
  g1[6] = (int)((uint32_t)((s0 >> 32) & 0xFFFFull) |
                (uint32_t)((s1 & 0xFFFFull) << 16));
  g1[7] = (int)(uint32_t)(s1 >> 16);
  int32x4 z4 = {0, 0, 0, 0};
  int32x8 z8 = {0, 0, 0, 0, 0, 0, 0, 0};
  // 6-arg form (clang-23 / therock-10.0 lane): groups 2/3 + extra group unused.
  __builtin_amdgcn_tensor_load_to_lds(g0, g1, z4, z4, z8, 0);
}
#endif

// ---------------- Kernel 1: emit[b][t][u] = enc[b,t,tgt]+dec[b,u,tgt] ----------------
__global__ __launch_bounds__(128)
void rnnt_emit_kernel(const float* __restrict__ enc, const float* __restrict__ dec,
                      const int* __restrict__ targets, float* __restrict__ emit) {
  const int t = blockIdx.x;
  const int b = blockIdx.y;
  const int u = threadIdx.x;
  if (u < U_ - 1) {
    const int tg = targets[(size_t)b * (U_ - 1) + u];
    const float e  = enc[(((size_t)b * T_) + t) * V_ + tg];
    const float dy = dec[(((size_t)b * U_) + u) * V_ + tg];
    emit[(((size_t)b * T_) + t) * EP + u] = e + dy;
  }
}

// ---------------- Kernel 2: anti-diagonal wavefront DP, one block per batch ----------------
__global__ __launch_bounds__(128)
void rnnt_dp_kernel(const float* __restrict__ enc, const float* __restrict__ dec,
                    const int* __restrict__ in_len, const int* __restrict__ tg_len,
                    const int* __restrict__ blank_p, const float* __restrict__ emit,
                    float* __restrict__ out) {
  __shared__ float s_encb[T_];
  __shared__ float s_decb[U_];
  __shared__ float sA[2][U_];

  const int b = blockIdx.x;
  const int u = threadIdx.x;                 // blockDim.x == 128 == U_
  const int blank = blank_p[0];

  const float* encb_g = enc + ((size_t)b * T_) * V_ + blank;
  const float* decb_g = dec + ((size_t)b * U_) * V_ + blank;

#if USE_TDM
  if (threadIdx.x < 32) {                    // wave 0 issues the two DMAs
    tdm_load_col_f32(encb_g, (uint32_t)(uintptr_t)(void*)&s_encb[0], T_, V_);
    tdm_load_col_f32(decb_g, (uint32_t)(uintptr_t)(void*)&s_decb[0], U_, V_);
    __builtin_amdgcn_s_wait_tensorcnt(0);
  }
#else
  for (int t = u; t < T_; t += 128) s_encb[t] = encb_g[(size_t)t * V_];
  s_decb[u] = decb_g[(size_t)u * V_];
#endif
  __syncthreads();

  const int   tlast  = in_len[b] - 1;        // in [256, 511]
  const int   tl     = tg_len[b];            // in [64, 127]
  const float decb_u = s_decb[u];
  const float decb0  = s_decb[0];

  // Per-thread emit column pointer: emit[b][t][u-1] = emrow[t*EP]
  const float* emrow = emit + ((size_t)b * T_) * EP + (u - 1);
  float em0 = 0.f, em_cur = 0.f;
  if (u >= 1) {
    em0    = emrow[0];                       // emit[0][u-1]  (t==0 boundary)
    em_cur = emrow[EP];                      // emit[1][u-1]  (first general use)
  }

  if (u == 0) sA[0][0] = 0.0f;               // alpha[0][0] = 0  (diagonal d=0)
  __syncthreads();

  int pp = 0;
  #pragma unroll 1
  for (int d = 1; d <= T_ + U_ - 2; ++d) {
    const float* Ap = sA[pp];
    float*       Ac = sA[pp ^ 1];
    const int t = d - u;
    if (t >= 0 && t < T_) {
      float a;
      if (u == 0) {                          // blank-only boundary (t = d >= 1)
        a = Ap[0] + s_encb[t - 1] + decb0;
      } else if (t == 0) {                   // emit-only boundary (u = d)
        a = Ap[u - 1] + em0;
      } else {                               // general cell: logaddexp
        const float x  = Ap[u]     + s_encb[t - 1] + decb_u;
        const float y  = Ap[u - 1] + em_cur;
        const float mx = fmaxf(x, y);
        const float mn = fminf(x, y);
        a = mx + log1pf(expf(mn - mx));
      }
      Ac[u] = a;
      if (t == tlast && u == tl)             // final log-prob for this batch
        out[b] = a + s_encb[t] + decb_u;
    }
    // Software pipeline: issue next diagonal's emit load before the barrier
    // so its L2 latency overlaps the barrier + other threads' work.
    if (u >= 1) {
      const int tn = t + 1;
      if (tn >= 1 && tn < T_) em_cur = emrow[(size_t)tn * EP];
    }
    __syncthreads();
    pp ^= 1;
  }
}

extern "C" void kernel_launch(void* const* d_in, const int* in_sizes, int n_in,
                              void* d_out, int out_size, void* d_ws, size_t ws_size,
                              hipStream_t stream) {
  (void)in_sizes; (void)n_in; (void)out_size; (void)ws_size;
  const float* enc     = (const float*)d_in[0];
  const float* dec     = (const float*)d_in[1];
  const int*   targets = (const int*)d_in[2];
  const int*   in_len  = (const int*)d_in[3];
  const int*   tg_len  = (const int*)d_in[4];
  const int*   blank   = (const int*)d_in[5];
  float*       out     = (float*)d_out;
  float*       emit    = (float*)d_ws;      // B*T*EP*4 = 8 MiB scratch

  rnnt_emit_kernel<<<dim3(T_, B_), 128, 0, stream>>>(enc, dec, targets, emit);
  rnnt_dp_kernel<<<B_, 128, 0, stream>>>(enc, dec, in_len, tg_len, blank, emit, out);
}